// Predictor_28999619182888
// MI455X (gfx1250) — compile-verified
//
#include <hip/hip_runtime.h>

#define C_CH 128
#define T_LEN 2048
#define B_N 8
#define NATOMS 3584
#define NEGS 0.2f

typedef __attribute__((ext_vector_type(2))) float v2f;
typedef __attribute__((ext_vector_type(8))) float v8f;

__device__ __forceinline__ v8f wmma4(v2f a, v2f b, v8f c) {
  // V_WMMA_F32_16X16X4_F32 : D = A(16x4,f32) x B(4x16,f32) + C(16x16,f32)
  return __builtin_amdgcn_wmma_f32_16x16x4_f32(false, a, false, b, (short)0, c,
                                               false, false);
}

__device__ __forceinline__ float lrelu(float v) { return v >= 0.f ? v : NEGS * v; }

// ---------------------------------------------------------------------------
// k0: fold posamp linear through reduce_w (rank-2 trick).
// ---------------------------------------------------------------------------
__global__ void k0_rank2(const float* __restrict__ rw, const float* __restrict__ rb,
                         const float* __restrict__ pw, const float* __restrict__ pb,
                         float* __restrict__ v012) {
  int c = threadIdx.x;
  const float* row = rw + c * 256 + 128;
  float s0 = 0.f, s1 = 0.f, s2 = 0.f;
  for (int j = 0; j < 128; ++j) {
    float w = row[j];
    s0 += w * pw[j * 2 + 0];
    s1 += w * pw[j * 2 + 1];
    s2 += w * pb[j];
  }
  v012[c] = s0;
  v012[128 + c] = s1;
  v012[256 + c] = s2 + rb[c];
}

// ---------------------------------------------------------------------------
// k0b: transpose dil_w [6][O][I][3] -> dwt [6][3][O][I] so WMMA B fragments
// become contiguous 8-byte loads in the hot conv loop.
// ---------------------------------------------------------------------------
__global__ void k0b_transpose(const float* __restrict__ dw, float* __restrict__ dwt) {
  int idx = blockIdx.x * 256 + threadIdx.x;  // 6*3*128*128 = 294912 total
  int k = idx & 127;
  int o = (idx >> 7) & 127;
  int tap = (idx >> 14) % 3;
  int i = idx / (3 << 14);
  dwt[idx] = dw[(((size_t)i * 128 + o) * 128 + k) * 3 + tap];
}

// ---------------------------------------------------------------------------
// k1: h0[t][c] = sum_k embed[idx_t][k]*rw[c][k] + pos*v0[c] + amp*v1[c] + v2[c]
// ---------------------------------------------------------------------------
__global__ void k1_embed_reduce(const float* __restrict__ x,
                                const float* __restrict__ embed,
                                const float* __restrict__ rw,
                                const float* __restrict__ v012,
                                float* __restrict__ h0) {
  __shared__ float sA[16 * 132];
  __shared__ float sPos[16];
  __shared__ float sAmp[16];
  const int t0 = blockIdx.x * 16;
  const int tid = threadIdx.x;

  {  // gather 16 embedding rows (16 threads x 8 floats per row)
    int row = tid >> 4, seg = tid & 15;
    int gt = t0 + row;
    int b = gt >> 11, t = gt & 2047;
    int idx = (int)x[(b * 4 + 0) * T_LEN + t];
    const float* src = embed + (size_t)idx * 128 + seg * 8;
    float4 p0 = *(const float4*)src;
    float4 p1 = *(const float4*)(src + 4);
    *(float4*)&sA[row * 132 + seg * 8] = p0;
    *(float4*)&sA[row * 132 + seg * 8 + 4] = p1;
    if (seg == 0) {
      sPos[row] = x[(b * 4 + 1) * T_LEN + t];
      sAmp[row] = x[(b * 4 + 2) * T_LEN + t];
    }
  }
  __syncthreads();

  const int lane = tid & 31;
  const int n0 = (tid >> 5) * 16;
  const int m = lane & 15, half = lane >> 4;
  const int n = n0 + (lane & 15);
  const float* brow = rw + n * 256;

  v8f acc = {0.f, 0.f, 0.f, 0.f, 0.f, 0.f, 0.f, 0.f};
  for (int kb = 0; kb < 128; kb += 4) {
    int k = kb + half * 2;
    v2f a;
    a.x = sA[m * 132 + k];
    a.y = sA[m * 132 + k + 1];
    v2f bb;
    bb.x = brow[k];
    bb.y = brow[k + 1];
    acc = wmma4(a, bb, acc);
  }
  const float v0 = v012[n], v1 = v012[128 + n], v2 = v012[256 + n];
#pragma unroll
  for (int r = 0; r < 8; ++r) {
    int mr = r + half * 8;
    h0[(size_t)(t0 + mr) * 128 + n] = acc[r] + sPos[mr] * v0 + sAmp[mr] * v1 + v2;
  }
}

// ---------------------------------------------------------------------------
// k2: one residual block.
//  stage:  three shifted 16x128 activation tiles -> LDS (masked, EXEC-full)
//  phase1: y = dilated k=3 conv + dil_b  -> LDS
//  phase2: h' = lrelu(one_w @ y + one_b + residual)   (residual = tap-1 tile)
// ---------------------------------------------------------------------------
__global__ void k2_resblock(const float* __restrict__ hin, float* __restrict__ hout,
                            const float* __restrict__ dwt,  // [3][128][128] this layer
                            const float* __restrict__ db,
                            const float* __restrict__ ow,   // [128][128]
                            const float* __restrict__ ob, int d) {
  __shared__ float sA[48 * 132];  // 3 tap tiles, stride-132 (conflict-free)
  __shared__ float sY[16 * 132];
  const int t0 = blockIdx.x * 16;
  const int tid = threadIdx.x;
  const int b = t0 >> 11;
  const int tloc = t0 & 2047;

  // cooperative staging: 48 rows x 16 segments of 8 floats = 768 units
#pragma unroll
  for (int it = 0; it < 3; ++it) {
    int u = tid + it * 256;
    int row = u >> 4, seg = u & 15;     // row 0..47
    int tap = row >> 4, r = row & 15;
    int trow = tloc + r + (tap - 1) * d;
    float msk = (trow >= 0 && trow < T_LEN) ? 1.f : 0.f;
    int trc = trow < 0 ? 0 : (trow >= T_LEN ? T_LEN - 1 : trow);
    const float* src = hin + (size_t)(b * T_LEN + trc) * 128 + seg * 8;
    float4 p0 = *(const float4*)src;
    float4 p1 = *(const float4*)(src + 4);
    p0.x *= msk; p0.y *= msk; p0.z *= msk; p0.w *= msk;
    p1.x *= msk; p1.y *= msk; p1.z *= msk; p1.w *= msk;
    *(float4*)&sA[row * 132 + seg * 8] = p0;
    *(float4*)&sA[row * 132 + seg * 8 + 4] = p1;
  }
  __syncthreads();

  const int lane = tid & 31;
  const int n0 = (tid >> 5) * 16;
  const int m = lane & 15, half = lane >> 4;
  const int n = n0 + (lane & 15);

  v8f acc = {0.f, 0.f, 0.f, 0.f, 0.f, 0.f, 0.f, 0.f};
#pragma unroll
  for (int tap = 0; tap < 3; ++tap) {
    const float* arow = sA + (tap * 16 + m) * 132;
    const float* bw = dwt + ((size_t)tap * 128 + n) * 128;  // contiguous in K
    for (int kb = 0; kb < 128; kb += 4) {
      int k = kb + half * 2;
      v2f a;
      a.x = arow[k];
      a.y = arow[k + 1];
      v2f bb;
      bb.x = bw[k];
      bb.y = bw[k + 1];
      acc = wmma4(a, bb, acc);
    }
  }
  const float dbn = db[n];
#pragma unroll
  for (int r = 0; r < 8; ++r) sY[(r + half * 8) * 132 + n] = acc[r] + dbn;
  __syncthreads();

  v8f acc2 = {0.f, 0.f, 0.f, 0.f, 0.f, 0.f, 0.f, 0.f};
  const float* bw2 = ow + (size_t)n * 128;
  for (int kb = 0; kb < 128; kb += 4) {
    int k = kb + half * 2;
    v2f a;
    a.x = sY[m * 132 + k];
    a.y = sY[m * 132 + k + 1];
    v2f bb;
    bb.x = bw2[k];
    bb.y = bw2[k + 1];
    acc2 = wmma4(a, bb, acc2);
  }
  const float obn = ob[n];
#pragma unroll
  for (int r = 0; r < 8; ++r) {
    int mr = r + half * 8;
    float res = sA[(16 + mr) * 132 + n];  // tap-1 tile == residual rows
    hout[(size_t)(t0 + mr) * 128 + n] = lrelu(acc2[r] + obn + res);
  }
}

// ---------------------------------------------------------------------------
// k3: only t = T-1 feeds the heads; both heads' hidden layers for 8 rows.
// ---------------------------------------------------------------------------
__global__ void k3_hidden(const float* __restrict__ hfin,
                          const float* __restrict__ aw1, const float* __restrict__ ab1,
                          const float* __restrict__ aw2, const float* __restrict__ ab2,
                          const float* __restrict__ pw1, const float* __restrict__ pb1,
                          const float* __restrict__ pw2, const float* __restrict__ pb2,
                          float* __restrict__ hidA, float* __restrict__ hidP) {
  __shared__ float sH[8 * 128];
  __shared__ float sU[2 * 8 * 128];
  const int tid = threadIdx.x;
  for (int i = tid; i < 8 * 128; i += 256) {
    int b = i >> 7, k = i & 127;
    sH[i] = hfin[(size_t)(b * T_LEN + (T_LEN - 1)) * 128 + k];
  }
  __syncthreads();
  for (int i = tid; i < 2 * 8 * 128; i += 256) {
    int head = i >> 10, rem = i & 1023, b = rem >> 7, c = rem & 127;
    const float* w = (head ? pw1 : aw1) + c * 128;
    float s = (head ? pb1 : ab1)[c];
    const float* hr = sH + b * 128;
    for (int k = 0; k < 128; ++k) s += hr[k] * w[k];
    sU[i] = lrelu(s);
  }
  __syncthreads();
  for (int i = tid; i < 2 * 8 * 128; i += 256) {
    int head = i >> 10, rem = i & 1023, b = rem >> 7, c = rem & 127;
    const float* w = (head ? pw2 : aw2) + c * 128;
    float s = (head ? pb2 : ab2)[c];
    const float* ur = sU + head * 1024 + b * 128;
    for (int k = 0; k < 128; ++k) s += ur[k] * w[k];
    float v = lrelu(s);
    if (head) hidP[rem] = v; else hidA[rem] = v;
  }
}

// ---------------------------------------------------------------------------
// k4: atom logits, M=8 (padded to 16) x N=3584, K=128 via WMMA. 224 tiles.
// ---------------------------------------------------------------------------
__global__ void k4_atom_out(const float* __restrict__ hidA,
                            const float* __restrict__ w3, const float* __restrict__ b3,
                            float* __restrict__ out) {
  const int tid = threadIdx.x;
  const int lane = tid & 31;
  const int tile = blockIdx.x * 8 + (tid >> 5);
  const int n0 = tile * 16;
  const int m = lane & 15, half = lane >> 4;
  const int n = n0 + (lane & 15);
  const float msk = (m < 8) ? 1.f : 0.f;
  const float* ar = hidA + (m & 7) * 128;
  const float* bw = w3 + (size_t)n * 128;

  v8f acc = {0.f, 0.f, 0.f, 0.f, 0.f, 0.f, 0.f, 0.f};
  for (int kb = 0; kb < 128; kb += 4) {
    int k = kb + half * 2;
    v2f a;
    a.x = ar[k] * msk;
    a.y = ar[k + 1] * msk;
    v2f bb;
    bb.x = bw[k];
    bb.y = bw[k + 1];
    acc = wmma4(a, bb, acc);
  }
  const float bias = b3[n];
#pragma unroll
  for (int r = 0; r < 8; ++r) {
    int mr = r + half * 8;
    if (mr < 8) out[(size_t)mr * NATOMS + n] = acc[r] + bias;
  }
}

__global__ void k5_pa_out(const float* __restrict__ hidP,
                          const float* __restrict__ w3, const float* __restrict__ b3,
                          float* __restrict__ out) {
  int tid = threadIdx.x;
  if (tid < 16) {
    int b = tid >> 1, j = tid & 1;
    float s = b3[j];
    for (int k = 0; k < 128; ++k) s += hidP[b * 128 + k] * w3[j * 128 + k];
    out[b * 2 + j] = s;
  }
}

// ---------------------------------------------------------------------------
extern "C" void kernel_launch(void* const* d_in, const int* in_sizes, int n_in,
                              void* d_out, int out_size, void* d_ws, size_t ws_size,
                              hipStream_t stream) {
  const float* x        = (const float*)d_in[0];
  const float* embed_w  = (const float*)d_in[1];
  const float* posamp_w = (const float*)d_in[2];
  const float* posamp_b = (const float*)d_in[3];
  const float* reduce_w = (const float*)d_in[4];
  const float* reduce_b = (const float*)d_in[5];
  const float* dil_w    = (const float*)d_in[6];
  const float* dil_b    = (const float*)d_in[7];
  const float* one_w    = (const float*)d_in[8];
  const float* one_b    = (const float*)d_in[9];
  const float* atom_w1  = (const float*)d_in[10];
  const float* atom_b1  = (const float*)d_in[11];
  const float* atom_w2  = (const float*)d_in[12];
  const float* atom_b2  = (const float*)d_in[13];
  const float* atom_w3  = (const float*)d_in[14];
  const float* atom_b3  = (const float*)d_in[15];
  const float* pa_w1    = (const float*)d_in[16];
  const float* pa_b1    = (const float*)d_in[17];
  const float* pa_w2    = (const float*)d_in[18];
  const float* pa_b2    = (const float*)d_in[19];
  const float* pa_w3    = (const float*)d_in[20];
  const float* pa_b3    = (const float*)d_in[21];
  float* out = (float*)d_out;

  float* ws   = (float*)d_ws;
  float* v012 = ws;                          // 384 floats
  float* hidA = ws + 384;                    // 1024
  float* hidP = ws + 384 + 1024;             // 1024
  float* dwt  = ws + 4096;                   // 6*3*128*128 = 294912
  float* hA   = ws + 4096 + 294912 + 2048;   // 16384*128 (aligned region)
  float* hB   = hA + (size_t)16384 * 128;

  k0_rank2<<<1, 128, 0, stream>>>(reduce_w, reduce_b, posamp_w, posamp_b, v012);
  k0b_transpose<<<1152, 256, 0, stream>>>(dil_w, dwt);
  k1_embed_reduce<<<1024, 256, 0, stream>>>(x, embed_w, reduce_w, v012, hA);

  const int dils[6] = {1, 3, 9, 27, 81, 1};
  const float* hin = hA;
  float* hout = hB;
  for (int i = 0; i < 6; ++i) {
    k2_resblock<<<1024, 256, 0, stream>>>(hin, hout,
                                          dwt + (size_t)i * 3 * 128 * 128,
                                          dil_b + i * 128,
                                          one_w + (size_t)i * 128 * 128,
                                          one_b + i * 128, dils[i]);
    const float* t = hout;
    hout = (float*)hin;
    hin = t;
  }
  // after 6 swaps, final activations are back in hA (== hin)
  k3_hidden<<<1, 256, 0, stream>>>(hin, atom_w1, atom_b1, atom_w2, atom_b2,
                                   pa_w1, pa_b1, pa_w2, pa_b2, hidA, hidP);
  k4_atom_out<<<28, 256, 0, stream>>>(hidA, atom_w3, atom_b3, out);
  k5_pa_out<<<1, 32, 0, stream>>>(hidP, pa_w3, pa_b3, out + 8 * NATOMS);
}